// PointGroup_32272384262310
// MI455X (gfx1250) — compile-verified
//
#include <hip/hip_runtime.h>
#include <hip/hip_bf16.h>

typedef float v2f __attribute__((ext_vector_type(2)));
typedef float v8f __attribute__((ext_vector_type(8)));

#define N_PTS 1048576
#define CCH   64
#define NCLS  20
#define LDSW  96   // 64 w1 cols + 32 padded seg cols (20 real + 12 zero)

static __device__ inline v8f wmma4(v2f a, v2f b, v8f c) {
    // V_WMMA_F32_16X16X4_F32 : D(16x16 f32) = A(16x4 f32) * B(4x16 f32) + C
    return __builtin_amdgcn_wmma_f32_16x16x4_f32(
        /*neg_a=*/false, a, /*neg_b=*/false, b,
        /*c_mod=*/(short)0, c, /*reuse_a=*/false, /*reuse_b=*/false);
}

// ---------------------------------------------------------------------------
// Kernel 0: zero the small ws accumulator region
// ws floats: [0,64) sum  [64,128) sumsq  [128,192) scale  [192,256) shift
//            [256] l1acc [257] cosacc [258] maskacc
// ---------------------------------------------------------------------------
__global__ void init_ws_k(float* ws) {
    int i = threadIdx.x;
    if (i < 260) ws[i] = 0.0f;
}

// ---------------------------------------------------------------------------
// Kernel 1: h = feat@w1 + b1 statistics (sum, sumsq per channel)  AND
//           logit = feat@wseg + bseg  written to d_out (non-temporal: the
//           80MB logit stream is never re-read; keep L2 free for feat).
// Tiles ascending: at kernel end L2 holds the ~192MB tail of feat.
// ---------------------------------------------------------------------------
__global__ __launch_bounds__(256) void pass1_k(
    const float* __restrict__ feat, const float* __restrict__ w1,
    const float* __restrict__ b1,   const float* __restrict__ wseg,
    const float* __restrict__ bseg, float* __restrict__ logit,
    float* __restrict__ wsum, float* __restrict__ wsumsq)
{
    __shared__ float sB[CCH * LDSW];     // 24 KB combined weight tile
    __shared__ float sAccS[CCH];
    __shared__ float sAccQ[CCH];

    const int tid = threadIdx.x;
    for (int i = tid; i < CCH * LDSW; i += 256) {
        int r = i / LDSW, c = i % LDSW;
        float v = 0.0f;
        if (c < CCH)            v = w1[r * CCH + c];
        else if (c < CCH + NCLS) v = wseg[r * NCLS + (c - CCH)];
        sB[i] = v;
    }
    if (tid < CCH) { sAccS[tid] = 0.0f; sAccQ[tid] = 0.0f; }
    __syncthreads();

    const int wave  = tid >> 5;
    const int lane  = tid & 31;
    const int lhalf = lane >> 4;     // 0: K even pair / rows 0-7, 1: K odd pair / rows 8-15
    const int l15   = lane & 15;

    float b1v[4], bsegv[2];
    #pragma unroll
    for (int nb = 0; nb < 4; nb++) b1v[nb] = b1[nb * 16 + l15];
    #pragma unroll
    for (int nb = 0; nb < 2; nb++) {
        int c = nb * 16 + l15;
        bsegv[nb] = (c < NCLS) ? bseg[c] : 0.0f;
    }

    float sum[4] = {0,0,0,0}, sq[4] = {0,0,0,0};

    const int ntiles = N_PTS / 128;  // 8 waves * 16 rows per block-tile
    for (int t = blockIdx.x; t < ntiles; t += gridDim.x) {
        const int m0 = t * 128 + wave * 16;

        // software prefetch of the next tile's A row (global_prefetch_b8)
        const int tn = t + gridDim.x;
        if (tn < ntiles)
            __builtin_prefetch(feat + (size_t)(tn * 128 + wave * 16 + l15) * CCH
                                    + 2 * lhalf, 0, 1);

        // A fragments: lane l15 -> row m0+l15 ; lhalf selects K pair
        v2f a[16];
        const float* arow = feat + (size_t)(m0 + l15) * CCH + 2 * lhalf;
        #pragma unroll
        for (int kb = 0; kb < 16; kb++)
            a[kb] = *(const v2f*)(arow + kb * 4);

        #pragma unroll
        for (int nb = 0; nb < 6; nb++) {
            v8f acc = {0,0,0,0,0,0,0,0};
            #pragma unroll
            for (int kb = 0; kb < 16; kb++) {
                const int kr = kb * 4 + 2 * lhalf;
                v2f b;
                b.x = sB[kr * LDSW + nb * 16 + l15];
                b.y = sB[(kr + 1) * LDSW + nb * 16 + l15];
                acc = wmma4(a[kb], b, acc);
            }
            if (nb < 4) {
                #pragma unroll
                for (int r = 0; r < 8; r++) {
                    float h = acc[r] + b1v[nb];
                    sum[nb] += h;
                    sq[nb]  += h * h;
                }
            } else {
                const int col = (nb - 4) * 16 + l15;
                if (col < NCLS) {
                    #pragma unroll
                    for (int r = 0; r < 8; r++) {
                        const int row = m0 + r + 8 * lhalf;
                        __builtin_nontemporal_store(acc[r] + bsegv[nb - 4],
                            &logit[(size_t)row * NCLS + col]);
                    }
                }
            }
        }
    }

    // lanes 16-31 hold the same channels for rows 8-15: fold into lanes 0-15
    #pragma unroll
    for (int nb = 0; nb < 4; nb++) {
        sum[nb] += __shfl_xor(sum[nb], 16, 32);
        sq[nb]  += __shfl_xor(sq[nb], 16, 32);
    }
    if (lhalf == 0) {
        #pragma unroll
        for (int nb = 0; nb < 4; nb++) {
            atomicAdd(&sAccS[nb * 16 + l15], sum[nb]);
            atomicAdd(&sAccQ[nb * 16 + l15], sq[nb]);
        }
    }
    __syncthreads();
    if (tid < CCH)           atomicAdd(&wsum[tid],          sAccS[tid]);
    else if (tid < 2 * CCH)  atomicAdd(&wsumsq[tid - CCH],  sAccQ[tid - CCH]);
}

// ---------------------------------------------------------------------------
// Kernel 2: finalize BN scale/shift
// ---------------------------------------------------------------------------
__global__ void bn_finalize_k(const float* wsum, const float* wsumsq,
                              const float* gamma, const float* beta,
                              float* scale, float* shift)
{
    int c = threadIdx.x;
    if (c < CCH) {
        const float inv_n = 1.0f / (float)N_PTS;
        float mean = wsum[c] * inv_n;
        float var  = wsumsq[c] * inv_n - mean * mean;
        float rstd = rsqrtf(var + 1e-3f);
        float sc   = gamma[c] * rstd;
        scale[c] = sc;
        shift[c] = beta[c] - mean * sc;
    }
}

// ---------------------------------------------------------------------------
// Kernel 3: recompute h via WMMA (DESCENDING tile order: first reads hit the
// feat tail still resident in 192MB L2 from pass1), fused BN+ReLU in
// D-fragments, then the 64->3 head via butterfly reduction; write bias_pred
// regular-temporal (12MB, re-read immediately by the loss kernel from L2).
// feat loads are non-temporal (last use).
// ---------------------------------------------------------------------------
__global__ __launch_bounds__(256) void pass2_k(
    const float* __restrict__ feat, const float* __restrict__ w1,
    const float* __restrict__ b1,   const float* __restrict__ w2,
    const float* __restrict__ b2,   const float* __restrict__ scale,
    const float* __restrict__ shift, float* __restrict__ bias_out)
{
    __shared__ float sW[CCH * CCH];  // 16 KB
    const int tid = threadIdx.x;
    for (int i = tid; i < CCH * CCH; i += 256) sW[i] = w1[i];
    __syncthreads();

    const int wave  = tid >> 5;
    const int lane  = tid & 31;
    const int lhalf = lane >> 4;
    const int l15   = lane & 15;

    float sc[4], sh[4], w2L[4][3];
    #pragma unroll
    for (int nb = 0; nb < 4; nb++) {
        const int c = nb * 16 + l15;
        const float s = scale[c];
        sc[nb] = s;
        sh[nb] = b1[c] * s + shift[c];   // fold b1 into the BN affine
        w2L[nb][0] = w2[c * 3 + 0];
        w2L[nb][1] = w2[c * 3 + 1];
        w2L[nb][2] = w2[c * 3 + 2];
    }
    const float b2v0 = b2[0], b2v1 = b2[1], b2v2 = b2[2];

    const int ntiles = N_PTS / 128;
    for (int t0 = blockIdx.x; t0 < ntiles; t0 += gridDim.x) {
        const int t  = ntiles - 1 - t0;          // reverse traversal for L2 reuse
        const int m0 = t * 128 + wave * 16;

        const int tn0 = t0 + gridDim.x;
        if (tn0 < ntiles) {
            const int tn = ntiles - 1 - tn0;
            __builtin_prefetch(feat + (size_t)(tn * 128 + wave * 16 + l15) * CCH
                                    + 2 * lhalf, 0, 1);
        }

        v2f a[16];
        const float* arow = feat + (size_t)(m0 + l15) * CCH + 2 * lhalf;
        #pragma unroll
        for (int kb = 0; kb < 16; kb++)
            a[kb] = __builtin_nontemporal_load((const v2f*)(arow + kb * 4));

        v8f hacc[4];
        #pragma unroll
        for (int nb = 0; nb < 4; nb++) {
            v8f acc = {0,0,0,0,0,0,0,0};
            #pragma unroll
            for (int kb = 0; kb < 16; kb++) {
                const int kr = kb * 4 + 2 * lhalf;
                v2f b;
                b.x = sW[kr * CCH + nb * 16 + l15];
                b.y = sW[(kr + 1) * CCH + nb * 16 + l15];
                acc = wmma4(a[kb], b, acc);
            }
            #pragma unroll
            for (int r = 0; r < 8; r++) {
                float h = acc[r] * sc[nb] + sh[nb];  // BN affine (b1 folded)
                acc[r] = fmaxf(h, 0.0f);             // ReLU
            }
            hacc[nb] = acc;
        }

        // bias head: per lane contributes its channel, reduce 16 lanes
        #pragma unroll
        for (int r = 0; r < 8; r++) {
            float s0 = 0.0f, s1 = 0.0f, s2 = 0.0f;
            #pragma unroll
            for (int nb = 0; nb < 4; nb++) {
                const float h = hacc[nb][r];
                s0 += h * w2L[nb][0];
                s1 += h * w2L[nb][1];
                s2 += h * w2L[nb][2];
            }
            #pragma unroll
            for (int m = 1; m < 16; m <<= 1) {
                s0 += __shfl_xor(s0, m, 32);
                s1 += __shfl_xor(s1, m, 32);
                s2 += __shfl_xor(s2, m, 32);
            }
            if (l15 == 0) {  // lanes 0 (rows 0-7) and 16 (rows 8-15) write
                const int row = m0 + r + 8 * lhalf;
                bias_out[(size_t)row * 3 + 0] = s0 + b2v0;
                bias_out[(size_t)row * 3 + 1] = s1 + b2v1;
                bias_out[(size_t)row * 3 + 2] = s2 + b2v2;
            }
        }
    }
}

// ---------------------------------------------------------------------------
// Kernel 4: per-point masked L1 + cosine loss partials
// ---------------------------------------------------------------------------
__global__ __launch_bounds__(256) void loss_partial_k(
    const float* __restrict__ coord, const float* __restrict__ bottom,
    const float* __restrict__ vstem, const int* __restrict__ inst,
    const float* __restrict__ bias_pred, float* __restrict__ accum)
{
    float l1 = 0.0f, cs = 0.0f, mk = 0.0f;
    for (int p = blockIdx.x * blockDim.x + threadIdx.x; p < N_PTS;
         p += gridDim.x * blockDim.x) {
        const size_t b = (size_t)p * 3;
        const float cx = coord[b + 0] - bottom[b + 0];
        const float cy = coord[b + 1] - bottom[b + 1];
        const float cz = coord[b + 2] - bottom[b + 2];
        const float vx = vstem[b + 0], vy = vstem[b + 1], vz = vstem[b + 2];
        const float proj = cx * vx + cy * vy + cz * vz;
        const float gx = proj * vx - cx;
        const float gy = proj * vy - cy;
        const float gz = proj * vz - cz;
        const float bx = bias_pred[b + 0];
        const float by = bias_pred[b + 1];
        const float bz = bias_pred[b + 2];
        const float m = (inst[p] != -1) ? 1.0f : 0.0f;
        const float d = fabsf(bx - gx) + fabsf(by - gy) + fabsf(bz - gz);
        const float nb_ = sqrtf(bx * bx + by * by + bz * bz) + 1e-8f;
        const float ng_ = sqrtf(gx * gx + gy * gy + gz * gz) + 1e-8f;
        const float cosv = -((bx * gx + by * gy + bz * gz) / (nb_ * ng_));
        l1 += d * m;
        cs += cosv * m;
        mk += m;
    }
    #pragma unroll
    for (int s = 1; s < 32; s <<= 1) {
        l1 += __shfl_xor(l1, s, 32);
        cs += __shfl_xor(cs, s, 32);
        mk += __shfl_xor(mk, s, 32);
    }
    if ((threadIdx.x & 31) == 0) {
        atomicAdd(&accum[0], l1);
        atomicAdd(&accum[1], cs);
        atomicAdd(&accum[2], mk);
    }
}

__global__ void loss_finalize_k(const float* accum, float* out_loss) {
    const float denom = accum[2] + 1e-8f;
    *out_loss = accum[0] / denom + accum[1] / denom;
}

// ---------------------------------------------------------------------------
// launcher
// inputs: 0 coord, 1 feat, 2 instance_bottom, 3 instance_stem_v, 4 instance,
//         5 w1, 6 b1, 7 gamma, 8 beta, 9 w2, 10 b2, 11 wseg, 12 bseg
// d_out:  [N*20 logit][N*3 bias_pred][1 loss]
// ---------------------------------------------------------------------------
extern "C" void kernel_launch(void* const* d_in, const int* in_sizes, int n_in,
                              void* d_out, int out_size, void* d_ws, size_t ws_size,
                              hipStream_t stream) {
    const float* coord  = (const float*)d_in[0];
    const float* feat   = (const float*)d_in[1];
    const float* bottom = (const float*)d_in[2];
    const float* vstem  = (const float*)d_in[3];
    const int*   inst   = (const int*)d_in[4];
    const float* w1     = (const float*)d_in[5];
    const float* b1     = (const float*)d_in[6];
    const float* gamma  = (const float*)d_in[7];
    const float* beta   = (const float*)d_in[8];
    const float* w2     = (const float*)d_in[9];
    const float* b2     = (const float*)d_in[10];
    const float* wseg   = (const float*)d_in[11];
    const float* bseg   = (const float*)d_in[12];

    float* out_logit = (float*)d_out;
    float* out_bias  = out_logit + (size_t)N_PTS * NCLS;
    float* out_loss  = out_logit + (size_t)N_PTS * (NCLS + 3);

    float* ws      = (float*)d_ws;
    float* wsum    = ws;
    float* wsumsq  = ws + 64;
    float* wscale  = ws + 128;
    float* wshift  = ws + 192;
    float* waccum  = ws + 256;

    init_ws_k<<<1, 256, 0, stream>>>(ws);
    pass1_k<<<2048, 256, 0, stream>>>(feat, w1, b1, wseg, bseg,
                                      out_logit, wsum, wsumsq);
    bn_finalize_k<<<1, 64, 0, stream>>>(wsum, wsumsq, gamma, beta, wscale, wshift);
    pass2_k<<<2048, 256, 0, stream>>>(feat, w1, b1, w2, b2, wscale, wshift, out_bias);
    loss_partial_k<<<2048, 256, 0, stream>>>(coord, bottom, vstem, inst,
                                             out_bias, waccum);
    loss_finalize_k<<<1, 1, 0, stream>>>(waccum, out_loss);
}